// Criterion_43001212567772
// MI455X (gfx1250) — compile-verified
//
#include <hip/hip_runtime.h>
#include <hip/hip_bf16.h>
#include <math.h>

typedef __attribute__((ext_vector_type(2))) float v2f;
typedef __attribute__((ext_vector_type(8))) float v8f;

#define B_N      4096
#define D_K      128
#define KSTEPS   (D_K / 4)      // 32 wmma_f32_16x16x4 steps per tile
#define NTILES   (B_N / 16)     // 256 tiles per dimension
#define POS_Wc   2.0f
#define NEG_Wc   40.0f
#define MARGINc  0.1f
#define THRESHc  0.5f
#define INF_F    __builtin_inff()

// LDS strip layout: sim[16][SIM_STRIDE] floats, padded so the two half-wave
// row groups (rows r and r+8) hit disjoint banks: 4100 % 64 = 4, 8*4 = 32.
#define SIM_STRIDE 4100
#define SIM_FLOATS (16 * SIM_STRIDE)            // 65,600 floats = 262,400 B
#define RED_FLOATS (8 * 16 * 16)                // per reduction array
#define SMEM_FLOATS (SIM_FLOATS + 2 * RED_FLOATS + 32)
#define SMEM_BYTES  (SMEM_FLOATS * sizeof(float))   // ~278.9 KB < 320 KB WGP LDS

#if defined(__has_builtin)
#if __has_builtin(__builtin_amdgcn_sched_barrier)
#define SCHED_FENCE() __builtin_amdgcn_sched_barrier(0)
#endif
#endif
#ifndef SCHED_FENCE
#define SCHED_FENCE()
#endif

// f32 16x4 WMMA A-fragment (and, by Gram symmetry, the 4x16 B-fragment):
// lane L, VGPR v <- X[rc][k0 + v + 2*(L>>4)]  ==> one global_load_b64 per lane.
__device__ __forceinline__ v2f load_frag(const float* __restrict__ X, int rc, int kstep, int half) {
  return *(const v2f*)(X + rc * D_K + kstep * 4 + half * 2);
}

// ---------------- Fused kernel: one 16-row strip per workgroup.
// Phase 1: GEMM strip -> LDS (single pass over X), per-row masked min/max.
// Phase 2: re-read strip from LDS, masked exponential sums, per-row loss.
// __launch_bounds__(256, 1): occupancy is LDS-bound at 1 WG/WGP anyway, so
// let the backend use the full VGPR budget (enables deep load multi-buffering).
__global__ void __launch_bounds__(256, 1) fused_loss_strip(
    const float* __restrict__ X, const int* __restrict__ labels,
    float* __restrict__ row_loss, float* __restrict__ validf) {
  extern __shared__ float smem[];
  float* sim    = smem;                       // [16][SIM_STRIDE]
  float* redA   = smem + SIM_FLOATS;          // [8][16][16]
  float* redB   = redA + RED_FLOATS;          // [8][16][16]
  float* thrMin = redB + RED_FLOATS;          // [16]
  float* thrMax = thrMin + 16;                // [16]

  const int lane  = threadIdx.x & 31;
  const int wave  = threadIdx.x >> 5;
  const int half  = lane >> 4;
  const int l15   = lane & 15;
  const int row0  = blockIdx.x * 16;
  const int rbase = half * 8;                 // C-layout: VGPR r holds row rbase + r

  // Preload this wave's A fragments for all 32 k-steps (64 VGPRs, phase 1 only)
  v2f afrag[KSTEPS];
#pragma unroll
  for (int s = 0; s < KSTEPS; ++s) afrag[s] = load_frag(X, row0 + l15, s, half);

  int rlab[8];
#pragma unroll
  for (int r = 0; r < 8; ++r) rlab[r] = labels[row0 + rbase + r];

  float pmin[8], nmax[8];
#pragma unroll
  for (int r = 0; r < 8; ++r) { pmin[r] = INF_F; nmax[r] = -INF_F; }

  // ---- Phase 1: single GEMM pass; stash sim strip in LDS, track min/max ----
  for (int ct = wave; ct < NTILES; ct += 8) {
    const int col = ct * 16 + l15;            // column this lane owns in the C tile
    const int clab = labels[col];             // issued alongside the batch below
    v2f bfrag[KSTEPS];                        // 64 VGPRs, 32 loads in flight
#pragma unroll
    for (int s = 0; s < KSTEPS; ++s) bfrag[s] = load_frag(X, col, s, half);
    // Keep all 32 loads issued ahead of the WMMA chain (don't sink/remat them).
    SCHED_FENCE();

    v8f acc = {};
#pragma unroll
    for (int s = 0; s < KSTEPS; ++s) {
      acc = __builtin_amdgcn_wmma_f32_16x16x4_f32(
          /*neg_a=*/false, afrag[s], /*neg_b=*/false, bfrag[s],
          /*c_mod=*/(short)0, acc, /*reuse_a=*/false, /*reuse_b=*/false);
    }
    SCHED_FENCE();

#pragma unroll
    for (int r = 0; r < 8; ++r) {
      const int   M  = row0 + rbase + r;
      const float sv = acc[r];
      const bool  eq = (clab == rlab[r]);
      pmin[r] = fminf(pmin[r], (eq && (col != M)) ? sv : INF_F);   // positives, not self
      nmax[r] = fmaxf(nmax[r], eq ? -INF_F : sv);                  // negatives
      sim[(rbase + r) * SIM_STRIDE + col] = sv;                    // bank-conflict-free
    }
  }

  // ---- Workgroup reduction of the hardest-example thresholds ----
#pragma unroll
  for (int r = 0; r < 8; ++r) {
    redA[(wave * 16 + rbase + r) * 16 + l15] = pmin[r];
    redB[(wave * 16 + rbase + r) * 16 + l15] = nmax[r];
  }
  __syncthreads();
  if (threadIdx.x < 16) {
    float m = INF_F, M = -INF_F;
    for (int w = 0; w < 8; ++w)
      for (int c = 0; c < 16; ++c) {
        m = fminf(m, redA[(w * 16 + threadIdx.x) * 16 + c]);
        M = fmaxf(M, redB[(w * 16 + threadIdx.x) * 16 + c]);
      }
    thrMin[threadIdx.x] = m;
    thrMax[threadIdx.x] = M;
  }
  __syncthreads();

  // ---- Phase 2: re-read strip from LDS, masked exponential sums ----
  float mp[8], mn[8], psum[8], nsum[8];
#pragma unroll
  for (int r = 0; r < 8; ++r) {
    mp[r] = thrMin[rbase + r];
    mn[r] = thrMax[rbase + r];
    psum[r] = 0.f;
    nsum[r] = 0.f;
  }

  for (int ct = wave; ct < NTILES; ct += 8) {
    const int col  = ct * 16 + l15;
    const int clab = labels[col];
#pragma unroll
    for (int r = 0; r < 8; ++r) {
      const int   M    = row0 + rbase + r;
      const float sv   = sim[(rbase + r) * SIM_STRIDE + col];
      const bool  eq   = (clab == rlab[r]);
      const bool  posk = eq && (col != M) && (sv - MARGINc < mn[r]);
      const bool  negk = (!eq)            && (sv + MARGINc > mp[r]);
      // exponents bounded in [-60, 20] -> kept terms never flush to 0
      psum[r] += posk ? expf(-POS_Wc * (sv - THRESHc)) : 0.f;
      nsum[r] += negk ? expf( NEG_Wc * (sv - THRESHc)) : 0.f;
    }
  }

  // ---- Workgroup reduction of the sums (redA/redB safely reusable here) ----
#pragma unroll
  for (int r = 0; r < 8; ++r) {
    redA[(wave * 16 + rbase + r) * 16 + l15] = psum[r];
    redB[(wave * 16 + rbase + r) * 16 + l15] = nsum[r];
  }
  __syncthreads();
  if (threadIdx.x < 16) {
    float ps = 0.f, ns = 0.f;
    for (int w = 0; w < 8; ++w)
      for (int c = 0; c < 16; ++c) {
        ps += redA[(w * 16 + threadIdx.x) * 16 + c];
        ns += redB[(w * 16 + threadIdx.x) * 16 + c];
      }
    const bool valid = (ps > 0.f) && (ns > 0.f);  // == any(pos_keep) && any(neg_keep)
    row_loss[row0 + threadIdx.x] = log1pf(ps) / POS_Wc + log1pf(ns) / NEG_Wc;
    validf[row0 + threadIdx.x]   = valid ? 1.f : 0.f;
  }
}

// ---------------- Deterministic valid-masked mean over 4096 rows
__global__ void __launch_bounds__(256) finalize_loss(
    const float* __restrict__ row_loss, const float* __restrict__ validf,
    float* __restrict__ out) {
  __shared__ float sl[256];
  __shared__ float sc[256];
  float ls = 0.f, cnt = 0.f;
  for (int i = threadIdx.x; i < B_N; i += 256) {
    const float v = validf[i];
    ls  += v * row_loss[i];
    cnt += v;
  }
  sl[threadIdx.x] = ls;
  sc[threadIdx.x] = cnt;
  __syncthreads();
  for (int off = 128; off > 0; off >>= 1) {
    if (threadIdx.x < off) {
      sl[threadIdx.x] += sl[threadIdx.x + off];
      sc[threadIdx.x] += sc[threadIdx.x + off];
    }
    __syncthreads();
  }
  if (threadIdx.x == 0) out[0] = sl[0] / fmaxf(sc[0], 1.0f);
}

extern "C" void kernel_launch(void* const* d_in, const int* in_sizes, int n_in,
                              void* d_out, int out_size, void* d_ws, size_t ws_size,
                              hipStream_t stream) {
  const float* X      = (const float*)d_in[0];  // [4096,128] f32, L2-normalized rows
  const int*   labels = (const int*)d_in[1];    // [4096] i32
  float*       out    = (float*)d_out;          // scalar f32

  float* ws       = (float*)d_ws;
  float* row_loss = ws;             // 4096
  float* validf   = ws + B_N;       // 4096   (32 KB of scratch)

  fused_loss_strip<<<NTILES, 256, SMEM_BYTES, stream>>>(X, labels, row_loss, validf);
  finalize_loss<<<1, 256, 0, stream>>>(row_loss, validf, out);
}